// SimpleModel_5901285064739
// MI455X (gfx1250) — compile-verified
//
#include <hip/hip_runtime.h>
#include <hip/hip_bf16.h>

// Shapes from the reference
#define VOCAB 100000
#define EDIM  128
#define BATCH 256
#define SEQ   512

typedef __attribute__((ext_vector_type(2))) float v2f;
typedef __attribute__((ext_vector_type(8))) float v8f;

// -------------------------------------------------------------------------
// Kernel 1: per-vocab-row dual projection as a GEMM via V_WMMA_F32_16X16X4_F32
//   qv[v] = dot(emb[v,:], Wq)        (D column N=0)
//   sv[v] = sum(emb[v,:])            (D column N=1, B column = ones)
// One 16-row vocab tile per wave, K=128 in 32 steps of 4.
// -------------------------------------------------------------------------
__global__ __launch_bounds__(128) void vocab_proj_kernel(
    const float* __restrict__ emb, const float* __restrict__ Wq,
    float* __restrict__ qv, float* __restrict__ sv) {
  // 4 waves/block; 16 rows x 132 floats (128 + 4 pad -> conflict-free b64 reads)
  __shared__ float tile[4 * 16 * 132];
  __shared__ float sWq[EDIM];

  const int t    = threadIdx.x;
  const int lane = t & 31;
  const int wave = t >> 5;
  if (t < EDIM) sWq[t] = Wq[t];

  const int tileIdx = blockIdx.x * 4 + wave;
  const int vbase   = tileIdx * 16;
  const bool active = (tileIdx < (VOCAB / 16));   // 100000 = 6250 * 16 exactly

  float* tw = &tile[wave * 16 * 132];
  if (active) {
    // Stage 16 rows x 128 f32 with coalesced float4 loads (512B per instruction)
    #pragma unroll 4
    for (int r = 0; r < 16; ++r) {
      const float4 v = *(const float4*)&emb[(size_t)(vbase + r) * EDIM + lane * 4];
      *(float4*)&tw[r * 132 + lane * 4] = v;
    }
  }
  __syncthreads();
  if (!active) return;

  // A-fragment addressing (16x4 f32): M = lane&15; lanes 0-15 hold K = {0,1},
  // lanes 16-31 hold K = {2,3}. B/D: N = lane&15, row-striped across lanes.
  const int m     = lane & 15;
  const int khalf = (lane >> 4) * 2;
  const int n     = m;                 // this lane's B/D column
  const float* arow  = &tw[m * 132 + khalf];
  const float* wqrow = &sWq[khalf];

  // Per-lane B-column selectors: col 0 = Wq slice, col 1 = ones, rest = 0.
  // b = wq * selq + sel1  -> unconditional ds_load + v_fma, no EXEC churn.
  const float selq = (n == 0) ? 1.0f : 0.0f;
  const float sel1 = (n == 1) ? 1.0f : 0.0f;

  v8f c = {};
  #pragma unroll 8
  for (int kk = 0; kk < EDIM; kk += 4) {
    v2f a;
    a.x = arow[kk];
    a.y = arow[kk + 1];
    const float wq0 = wqrow[kk];       // lanes in each half broadcast-read LDS
    const float wq1 = wqrow[kk + 1];
    v2f b;
    b.x = fmaf(wq0, selq, sel1);
    b.y = fmaf(wq1, selq, sel1);
    c = __builtin_amdgcn_wmma_f32_16x16x4_f32(
        /*neg_a=*/false, a, /*neg_b=*/false, b,
        /*c_mod=*/(short)0, c, /*reuse_a=*/false, /*reuse_b=*/false);
  }

  // D layout: VGPR j -> M=j (lanes 0-15, N=lane) / M=j+8 (lanes 16-31, N=lane-16)
  const int col = lane & 15;
  const int mb  = (lane >> 4) * 8;
  if (col == 0) {
    #pragma unroll
    for (int j = 0; j < 8; ++j) qv[vbase + mb + j] = c[j];
  } else if (col == 1) {
    #pragma unroll
    for (int j = 0; j < 8; ++j) sv[vbase + mb + j] = c[j];
  }
}

// -------------------------------------------------------------------------
// Kernel 2: log_p[b,s] = (qv[x]+bq)*sv[x]; stable softmax over S per row.
// One 256-thread block per batch row, 2 tokens per thread.
// -------------------------------------------------------------------------
__global__ __launch_bounds__(256) void softmax_kernel(
    const int* __restrict__ x, const float* __restrict__ qv,
    const float* __restrict__ sv, const float* __restrict__ bq,
    float* __restrict__ p) {
  __shared__ float red[256];
  const int b = blockIdx.x;
  const int t = threadIdx.x;
  const float bqv = bq[0];

  const int i0 = x[b * SEQ + t];
  const int i1 = x[b * SEQ + t + 256];
  const float l0 = (qv[i0] + bqv) * sv[i0];
  const float l1 = (qv[i1] + bqv) * sv[i1];

  // block max
  red[t] = fmaxf(l0, l1);
  __syncthreads();
  for (int off = 128; off > 0; off >>= 1) {
    if (t < off) red[t] = fmaxf(red[t], red[t + off]);
    __syncthreads();
  }
  const float mx = red[0];
  __syncthreads();

  const float e0 = __expf(l0 - mx);
  const float e1 = __expf(l1 - mx);
  red[t] = e0 + e1;
  __syncthreads();
  for (int off = 128; off > 0; off >>= 1) {
    if (t < off) red[t] += red[t + off];
    __syncthreads();
  }
  const float inv = 1.0f / red[0];

  p[b * SEQ + t]       = e0 * inv;
  p[b * SEQ + t + 256] = e1 * inv;
}

// -------------------------------------------------------------------------
// Kernel 3: t_hat[b,e] = sum_s p[b,s]*emb[x[b,s],e] (coalesced 512B row reads,
// mostly L2 hits), fused with out[b,:] = relu(t_hat @ Wm + bm).
// One 128-thread block per batch row; thread = embedding lane.
// -------------------------------------------------------------------------
__global__ __launch_bounds__(128) void pool_head_kernel(
    const int* __restrict__ x, const float* __restrict__ p,
    const float* __restrict__ emb, const float* __restrict__ Wm,
    const float* __restrict__ bm, float* __restrict__ out) {
  __shared__ int   sx[SEQ];
  __shared__ float sp[SEQ];
  __shared__ float red[128];
  const int b = blockIdx.x;
  const int t = threadIdx.x;

  for (int s = t; s < SEQ; s += 128) {
    sx[s] = x[b * SEQ + s];
    sp[s] = p[b * SEQ + s];
  }
  __syncthreads();

  float acc = 0.0f;
  #pragma unroll 4
  for (int s = 0; s < SEQ; ++s) {
    acc = fmaf(sp[s], emb[(size_t)sx[s] * EDIM + t], acc);
  }

  const float w0 = Wm[t * 2 + 0];
  const float w1 = Wm[t * 2 + 1];

  red[t] = acc * w0;
  __syncthreads();
  for (int off = 64; off > 0; off >>= 1) {
    if (t < off) red[t] += red[t + off];
    __syncthreads();
  }
  if (t == 0) out[b * 2 + 0] = fmaxf(red[0] + bm[0], 0.0f);
  __syncthreads();

  red[t] = acc * w1;
  __syncthreads();
  for (int off = 64; off > 0; off >>= 1) {
    if (t < off) red[t] += red[t + off];
    __syncthreads();
  }
  if (t == 0) out[b * 2 + 1] = fmaxf(red[0] + bm[1], 0.0f);
}

extern "C" void kernel_launch(void* const* d_in, const int* in_sizes, int n_in,
                              void* d_out, int out_size, void* d_ws, size_t ws_size,
                              hipStream_t stream) {
  const int*   x   = (const int*)d_in[0];    // [B,S] int32
  const float* emb = (const float*)d_in[1];  // [VOCAB,E]
  const float* Wq  = (const float*)d_in[2];  // [E,1]
  const float* bq  = (const float*)d_in[3];  // [1]
  const float* Wm  = (const float*)d_in[4];  // [E,2]
  const float* bm  = (const float*)d_in[5];  // [2]
  float* out = (float*)d_out;                // [B,2]

  float* ws = (float*)d_ws;
  float* qv = ws;                 // [VOCAB]
  float* sv = ws + VOCAB;         // [VOCAB]
  float* p  = ws + 2 * VOCAB;     // [B,S]

  const int tiles   = VOCAB / 16;           // 6250
  const int blocks1 = (tiles + 3) / 4;      // 4 waves (tiles) per block

  vocab_proj_kernel<<<blocks1, 128, 0, stream>>>(emb, Wq, qv, sv);
  softmax_kernel<<<BATCH, 256, 0, stream>>>(x, qv, sv, bq, p);
  pool_head_kernel<<<BATCH, 128, 0, stream>>>(x, p, emb, Wm, bm, out);
}